// Attention_83932250898836
// MI455X (gfx1250) — compile-verified
//
#include <hip/hip_runtime.h>
#include <math.h>

#define BATCH 2
#define SEQ   2048
#define NCH   1024
#define NH    16
#define HD    64

typedef __attribute__((ext_vector_type(2))) float v2f;
typedef __attribute__((ext_vector_type(8))) float v8f;

static __device__ __forceinline__ v8f wmma_f32(v2f a, v2f b, v8f c) {
  // D = A(16x4,f32) * B(4x16,f32) + C(16x16,f32)
  return __builtin_amdgcn_wmma_f32_16x16x4_f32(false, a, false, b, (short)0, c,
                                               false, false);
}

static __device__ __forceinline__ v2f ld2(const float* p) {
  return *(const v2f*)p;  // 8B-aligned by construction
}

// ---------------------------------------------------------------------------
// GEMM core: one wave computes a 16x64 tile of  Y = A(MxK) @ W(NxK)^T
// A row-major lda=K ; W row-major ldw=K (so B_wmma[k][n] = W[n][k])
// ---------------------------------------------------------------------------
static __device__ __forceinline__ void gemm_tile_16x64(
    const float* __restrict__ A, const float* __restrict__ W, int K,
    int m0, int n0, int lane, v8f acc[4]) {
  const int half = lane >> 4;
  const int l    = lane & 15;
  const float* arow = A + (size_t)(m0 + l) * K;
  const float* w0 = W + (size_t)(n0 + 0  + l) * K;
  const float* w1 = W + (size_t)(n0 + 16 + l) * K;
  const float* w2 = W + (size_t)(n0 + 32 + l) * K;
  const float* w3 = W + (size_t)(n0 + 48 + l) * K;
  const int ko = 2 * half;
  for (int k0 = 0; k0 < K; k0 += 4) {
    v2f a = ld2(arow + k0 + ko);
    acc[0] = wmma_f32(a, ld2(w0 + k0 + ko), acc[0]);
    acc[1] = wmma_f32(a, ld2(w1 + k0 + ko), acc[1]);
    acc[2] = wmma_f32(a, ld2(w2 + k0 + ko), acc[2]);
    acc[3] = wmma_f32(a, ld2(w3 + k0 + ko), acc[3]);
  }
}

// ---------------------------------------------------------------------------
// Kernel 1: QKV projection. Y[4096,3072] = x @ qkv_w^T + b, scattered into
// Q/K/V buffers laid out [B, H, T, hd].
// ---------------------------------------------------------------------------
__global__ __launch_bounds__(128) void qkv_gemm_kernel(
    const float* __restrict__ x, const float* __restrict__ w,
    const float* __restrict__ bias, float* __restrict__ qb,
    float* __restrict__ kb, float* __restrict__ vb) {
  const int lane = threadIdx.x & 31;
  const int half = lane >> 4;
  const int l    = lane & 15;
  const int wave = blockIdx.x * 4 + (threadIdx.x >> 5);
  const int MT = (BATCH * SEQ) / 16;  // 256
  const int mt = wave % MT;
  const int nt = wave / MT;           // 0..47
  const int m0 = mt * 16;
  const int n0 = nt * 64;

  const v8f z = {0.f, 0.f, 0.f, 0.f, 0.f, 0.f, 0.f, 0.f};
  v8f acc[4] = {z, z, z, z};
  gemm_tile_16x64(x, w, NCH, m0, n0, lane, acc);

  const int which = n0 / NCH;          // 0=q 1=k 2=v (64-wide tile never splits)
  const int h     = (n0 % NCH) / HD;   // head (64-wide tile stays in one head)
  const int b     = m0 / SEQ;
  const int t0    = m0 % SEQ;
  float* dst = (which == 0 ? qb : which == 1 ? kb : vb) +
               ((size_t)(b * NH + h) * SEQ) * HD;
#pragma unroll
  for (int j = 0; j < 4; ++j) {
    const float bv = bias[n0 + 16 * j + l];
#pragma unroll
    for (int i = 0; i < 8; ++i) {
      const int t = t0 + i + 8 * half;
      dst[(size_t)t * HD + 16 * j + l] = acc[j][i] + bv;
    }
  }
}

// ---------------------------------------------------------------------------
// Kernel 2: causal flash attention. One wave = 16 query rows of one (b,h).
// Q/K/V are [B,H,T,hd]; output is [B,T,C] (c = h*hd + d).
// ---------------------------------------------------------------------------
__global__ __launch_bounds__(128) void attn_kernel(
    const float* __restrict__ qbuf, const float* __restrict__ kbuf,
    const float* __restrict__ vbuf, float* __restrict__ obuf) {
  __shared__ float pshare[4][16 * 18];  // per-wave P tile, stride 18 (8B align)
  const int lane  = threadIdx.x & 31;
  const int half  = lane >> 4;
  const int l     = lane & 15;
  const int wslot = threadIdx.x >> 5;
  const int wave  = blockIdx.x * 4 + wslot;
  const int QT = SEQ / 16;             // 128 query tiles
  const int bh = wave / QT;            // 0..31
  const int qt = wave % QT;
  const int q0 = qt * 16;
  const int b = bh / NH;
  const int h = bh % NH;

  const float* Qp = qbuf + (size_t)bh * SEQ * HD;
  const float* Kp = kbuf + (size_t)bh * SEQ * HD;
  const float* Vp = vbuf + (size_t)bh * SEQ * HD;
  float* pb = pshare[wslot];
  const int ko = 2 * half;

  // Preload Q fragments for all 16 k-steps over hd=64.
  v2f qa[16];
#pragma unroll
  for (int s = 0; s < 16; ++s)
    qa[s] = ld2(Qp + (size_t)(q0 + l) * HD + 4 * s + ko);

  const v8f z = {0.f, 0.f, 0.f, 0.f, 0.f, 0.f, 0.f, 0.f};
  v8f o[4] = {z, z, z, z};
  float m[8], lsum[8];
#pragma unroll
  for (int i = 0; i < 8; ++i) { m[i] = -INFINITY; lsum[i] = 0.f; }

  const float scale = 0.125f;  // hd^-0.5

  for (int kk0 = 0; kk0 <= q0; kk0 += 16) {
    // S = Q @ K^T : B_wmma[k][n] = K[key=kk0+n][d=k]
    v8f s = z;
#pragma unroll
    for (int st = 0; st < 16; ++st) {
      v2f kf = ld2(Kp + (size_t)(kk0 + l) * HD + 4 * st + ko);
      s = wmma_f32(qa[st], kf, s);
    }
    const bool diag = (kk0 == q0);
#pragma unroll
    for (int i = 0; i < 8; ++i) {
      const int qi = i + 8 * half;              // row within tile
      float sv = s[i] * scale;
      if (diag && (l > qi)) sv = -INFINITY;     // causal mask on diagonal tile
      // row max over the 16 lanes holding this row
      float rm = sv;
      rm = fmaxf(rm, __shfl_xor(rm, 1, 32));
      rm = fmaxf(rm, __shfl_xor(rm, 2, 32));
      rm = fmaxf(rm, __shfl_xor(rm, 4, 32));
      rm = fmaxf(rm, __shfl_xor(rm, 8, 32));
      const float mnew  = fmaxf(m[i], rm);
      const float alpha = __expf(m[i] - mnew);
      const float p     = __expf(sv - mnew);
      float rs = p;
      rs += __shfl_xor(rs, 1, 32);
      rs += __shfl_xor(rs, 2, 32);
      rs += __shfl_xor(rs, 4, 32);
      rs += __shfl_xor(rs, 8, 32);
      lsum[i] = lsum[i] * alpha + rs;
      m[i] = mnew;
#pragma unroll
      for (int j = 0; j < 4; ++j) o[j][i] *= alpha;
      pb[(i + 8 * half) * 18 + l] = p;          // stage P for transpose
    }
    __threadfence_block();  // order LDS stores -> loads (same wave, dscnt)
    // O += P @ V : A = P (16x16), B = V rows [kk0..kk0+15] x hd
#pragma unroll
    for (int st = 0; st < 4; ++st) {
      v2f pa = ld2(pb + l * 18 + 4 * st + ko);
      const float* vrow = Vp + (size_t)(kk0 + 4 * st + ko) * HD;
#pragma unroll
      for (int j = 0; j < 4; ++j) {
        v2f bv;
        bv.x = vrow[16 * j + l];
        bv.y = vrow[HD + 16 * j + l];
        o[j] = wmma_f32(pa, bv, o[j]);
      }
    }
    __threadfence_block();  // keep next iteration's stores after these loads
  }

  // Normalize and write O -> [B,T,C]
#pragma unroll
  for (int j = 0; j < 4; ++j) {
#pragma unroll
    for (int i = 0; i < 8; ++i) {
      const int t = q0 + i + 8 * half;
      obuf[((size_t)b * SEQ + t) * NCH + h * HD + 16 * j + l] =
          o[j][i] / lsum[i];
    }
  }
}

// ---------------------------------------------------------------------------
// Kernel 3: output projection. out = attn @ proj_w^T + proj_b
// ---------------------------------------------------------------------------
__global__ __launch_bounds__(128) void proj_gemm_kernel(
    const float* __restrict__ a, const float* __restrict__ w,
    const float* __restrict__ bias, float* __restrict__ out) {
  const int lane = threadIdx.x & 31;
  const int half = lane >> 4;
  const int l    = lane & 15;
  const int wave = blockIdx.x * 4 + (threadIdx.x >> 5);
  const int MT = (BATCH * SEQ) / 16;  // 256
  const int mt = wave % MT;
  const int nt = wave / MT;           // 0..15
  const int m0 = mt * 16;
  const int n0 = nt * 64;

  const v8f z = {0.f, 0.f, 0.f, 0.f, 0.f, 0.f, 0.f, 0.f};
  v8f acc[4] = {z, z, z, z};
  gemm_tile_16x64(a, w, NCH, m0, n0, lane, acc);

#pragma unroll
  for (int j = 0; j < 4; ++j) {
    const float bv = bias[n0 + 16 * j + l];
#pragma unroll
    for (int i = 0; i < 8; ++i) {
      const int mrow = m0 + i + 8 * half;
      out[(size_t)mrow * NCH + n0 + 16 * j + l] = acc[j][i] + bv;
    }
  }
}

// ---------------------------------------------------------------------------
extern "C" void kernel_launch(void* const* d_in, const int* in_sizes, int n_in,
                              void* d_out, int out_size, void* d_ws,
                              size_t ws_size, hipStream_t stream) {
  const float* x      = (const float*)d_in[0];
  const float* qkv_w  = (const float*)d_in[1];
  const float* qkv_b  = (const float*)d_in[2];
  const float* proj_w = (const float*)d_in[3];
  const float* proj_b = (const float*)d_in[4];
  float* out = (float*)d_out;

  const size_t per = (size_t)BATCH * NH * SEQ * HD;  // 4,194,304 floats
  float* qb = (float*)d_ws;
  float* kb = qb + per;
  float* vb = kb + per;
  float* ab = vb + per;  // attention output [B,T,C], also `per` floats

  // QKV: 256 M-tiles * 48 N-tiles = 12288 waves / 4 per block
  qkv_gemm_kernel<<<3072, 128, 0, stream>>>(x, qkv_w, qkv_b, qb, kb, vb);
  // Attention: 2*16*128 = 4096 waves / 4 per block
  attn_kernel<<<1024, 128, 0, stream>>>(qb, kb, vb, ab);
  // Proj: 256 * 16 = 4096 waves / 4 per block
  proj_gemm_kernel<<<1024, 128, 0, stream>>>(ab, proj_w, proj_b, out);
}